// SpanFeaure_45835890983234
// MI455X (gfx1250) — compile-verified
//
#include <hip/hip_runtime.h>
#include <hip/hip_bf16.h>
#include <math.h>

typedef float v2f __attribute__((ext_vector_type(2)));
typedef float v8f __attribute__((ext_vector_type(8)));

// ---------------------------------------------------------------------------
// Async global->LDS staging (gfx1250 ASYNCcnt path), guarded so the file
// still compiles (falling back to sync staging) if the toolchain lacks the
// builtins. Signatures verified from compiler diagnostics:
//   b64: (v2i32 AS1* src, v2i32 AS3* dst, imm offset, imm cpol)
//   b32: (int   AS1* src, int   AS3* dst, imm offset, imm cpol)
// ---------------------------------------------------------------------------
#if defined(__has_builtin)
#if __has_builtin(__builtin_amdgcn_global_load_async_to_lds_b64) && \
    __has_builtin(__builtin_amdgcn_global_load_async_to_lds_b32) && \
    __has_builtin(__builtin_amdgcn_s_wait_asynccnt)
#define USE_ASYNC_STAGE 1
#endif
#endif
#ifndef USE_ASYNC_STAGE
#define USE_ASYNC_STAGE 0
#endif

#if USE_ASYNC_STAGE
typedef int v2i32 __attribute__((vector_size(8)));
typedef __attribute__((address_space(1))) v2i32 gv2i32;   // global int2
typedef __attribute__((address_space(3))) v2i32 lv2i32;   // LDS int2
typedef __attribute__((address_space(1))) int   gi32;     // global int
typedef __attribute__((address_space(3))) int   li32;     // LDS int

__device__ __forceinline__ void async_g2l_b64(const void* g, void* l) {
    __builtin_amdgcn_global_load_async_to_lds_b64((gv2i32*)g, (lv2i32*)l, 0, 0);
}
__device__ __forceinline__ void async_g2l_b32(const void* g, void* l) {
    __builtin_amdgcn_global_load_async_to_lds_b32((gi32*)g, (li32*)l, 0, 0);
}
#endif

// ---------------------------------------------------------------------------
// fp32 WMMA: D = A(16x4) * B(4x16) + C(16x16), exact fp32 per CDNA5 ISA
// ---------------------------------------------------------------------------
__device__ __forceinline__ v8f wmma_f32x4(v2f a, v2f b, v8f c) {
    // 8 args: (neg_a, A, neg_b, B, c_mod, C, reuse_a, reuse_b)
    return __builtin_amdgcn_wmma_f32_16x16x4_f32(
        false, a, false, b, (short)0, c, false, false);
}

// ---------------------------------------------------------------------------
// Tiled GEMM:  C[M,N] = scale * (A[M,K] x B) + bias
//   BT=true : B is W[N,K] row-major (computes A @ W^T, torch Linear style)
//   BT=false: B is [K,N] row-major  (plain A @ B)
//
// Block: 128 threads = 4 waves. Block tile 128(M) x 64(N), Kc=32.
// Wave w owns rows [w*32, w*32+32): 2 M-frags x 4 N-frags = 8 v8f accums.
// LDS holds K-pairs as float2 so every fragment load is one ds_load_b64:
//   AsP[kp][m] = { A[m][2kp], A[m][2kp+1] }   (fragment-ready layout)
// Row pad of 4 float2 -> row stride = 264/136 floats = 8 (mod 64) banks.
// With USE_ASYNC_STAGE: double-buffered async copies overlap the next
// K-chunk's HBM traffic with the current chunk's WMMAs.
// ---------------------------------------------------------------------------
template <bool BT>
__global__ __launch_bounds__(128)
void gemm128_wmma(const float* __restrict__ A, const float* __restrict__ B,
                  const float* __restrict__ bias, float* __restrict__ C,
                  int M, int N, int K, float scale) {
#if USE_ASYNC_STAGE
    __shared__ float2 AsP[2][16][132];   // double buffered
    __shared__ float2 BsP[2][16][68];
#else
    __shared__ float2 AsP[1][16][132];
    __shared__ float2 BsP[1][16][68];
#endif

    const int n0 = blockIdx.x * 64;
    const int m0 = blockIdx.y * 128;

    const int tid   = threadIdx.x;
    const int lane  = tid & 31;
    const int wave  = tid >> 5;             // 0..3
    const int mrow  = lane & 15;
    const int nlane = lane & 15;
    const int hi    = (lane < 16) ? 0 : 1;  // pair-row select for K halves
    const int mw    = wave * 32;            // wave's M base inside block

    v8f acc[2][4];
    const v8f vzero = {0.f, 0.f, 0.f, 0.f, 0.f, 0.f, 0.f, 0.f};
#pragma unroll
    for (int mt = 0; mt < 2; ++mt)
#pragma unroll
        for (int nt = 0; nt < 4; ++nt) acc[mt][nt] = vzero;

    // one K-chunk of WMMAs out of LDS buffer `buf`
    auto computeChunk = [&](int buf) {
#pragma unroll
        for (int kk = 0; kk < 32; kk += 4) {
            const int kpr = (kk >> 1) + hi;  // lanes 0-15: K=kk,kk+1; 16-31: +2,+3
            v2f a[2], b[4];
            {
                float2 t0 = AsP[buf][kpr][mw + mrow];
                float2 t1 = AsP[buf][kpr][mw + 16 + mrow];
                a[0].x = t0.x; a[0].y = t0.y;
                a[1].x = t1.x; a[1].y = t1.y;
            }
#pragma unroll
            for (int nt = 0; nt < 4; ++nt) {
                float2 t = BsP[buf][kpr][nt * 16 + nlane];
                b[nt].x = t.x; b[nt].y = t.y;
            }
#pragma unroll
            for (int mt = 0; mt < 2; ++mt)
#pragma unroll
                for (int nt = 0; nt < 4; ++nt)
                    acc[mt][nt] = wmma_f32x4(a[mt], b[nt], acc[mt][nt]);
        }
    };

#if USE_ASYNC_STAGE
    // ---- async double-buffered pipeline -----------------------------------
    auto stage = [&](int buf, int k0) {
#pragma unroll
        for (int i = tid; i < 128 * 16; i += 128) {        // A: 16 instr/wave
            int kp = i & 15, mm = i >> 4;
            async_g2l_b64(&A[(size_t)(m0 + mm) * K + k0 + 2 * kp],
                          &AsP[buf][kp][mm]);
        }
        if (BT) {
#pragma unroll
            for (int i = tid; i < 64 * 16; i += 128) {     // B: 8 instr/wave
                int kp = i & 15, nn = i >> 4;
                async_g2l_b64(&B[(size_t)(n0 + nn) * K + k0 + 2 * kp],
                              &BsP[buf][kp][nn]);
            }
        } else {
#pragma unroll
            for (int i = tid; i < 64 * 16; i += 128) {     // B: 16 instr/wave
                int nn = i & 63, kp = i >> 6;
                float* cell = (float*)&BsP[buf][kp][nn];
                async_g2l_b32(&B[(size_t)(k0 + 2 * kp)     * N + n0 + nn], cell);
                async_g2l_b32(&B[(size_t)(k0 + 2 * kp + 1) * N + n0 + nn], cell + 1);
            }
        }
    };
    constexpr short NSTAGE = BT ? 24 : 48;  // wait threshold; a stage issues 24 (BT) / 32 (!BT) async instrs per wave

    stage(0, 0);
    for (int k0 = 0; k0 < K; k0 += 32) {
        const int buf = (k0 >> 5) & 1;
        if (k0 + 32 < K) {
            stage(buf ^ 1, k0 + 32);                       // next chunk in flight
            __builtin_amdgcn_s_wait_asynccnt(NSTAGE);      // current chunk landed for BT only (!BT: waits to 48 of up to 64)
        } else {
            __builtin_amdgcn_s_wait_asynccnt(0);
        }
        __syncthreads();      // all waves' copies visible
        computeChunk(buf);
        __syncthreads();      // buf^1 reads done before it is overwritten
    }
#else
    // ---- synchronous staging fallback -------------------------------------
    for (int k0 = 0; k0 < K; k0 += 32) {
        if (k0 + 32 < K) {   // lowers to global_prefetch_b8
            __builtin_prefetch(&A[(size_t)(m0 + tid) * K + k0 + 32], 0, 1);
            if (BT) __builtin_prefetch(&B[(size_t)(n0 + (tid & 63)) * K + k0 + 32], 0, 1);
            else    __builtin_prefetch(&B[(size_t)(k0 + 32 + (tid >> 6)) * N + n0], 0, 1);
        }
        __syncthreads();
#pragma unroll
        for (int i = tid; i < 128 * 16; i += 128) {
            int kp = i & 15, mm = i >> 4;
            AsP[0][kp][mm] = *(const float2*)&A[(size_t)(m0 + mm) * K + k0 + 2 * kp];
        }
        if (BT) {
#pragma unroll
            for (int i = tid; i < 64 * 16; i += 128) {
                int kp = i & 15, nn = i >> 4;
                BsP[0][kp][nn] = *(const float2*)&B[(size_t)(n0 + nn) * K + k0 + 2 * kp];
            }
        } else {
#pragma unroll
            for (int i = tid; i < 64 * 16; i += 128) {
                int nn = i & 63, kp = i >> 6;
                float2 t;
                t.x = B[(size_t)(k0 + 2 * kp)     * N + n0 + nn];
                t.y = B[(size_t)(k0 + 2 * kp + 1) * N + n0 + nn];
                BsP[0][kp][nn] = t;
            }
        }
        __syncthreads();
        computeChunk(0);
    }
#endif

    // --- store: lane L, VGPR r holds M = r + (L<16?0:8), N = L&15
    const int rofs = (lane < 16) ? 0 : 8;
#pragma unroll
    for (int nt = 0; nt < 4; ++nt) {
        const int col = n0 + nt * 16 + nlane;
        const float bb = bias ? bias[col] : 0.0f;
#pragma unroll
        for (int mt = 0; mt < 2; ++mt) {
#pragma unroll
            for (int r = 0; r < 8; ++r) {
                const int row = m0 + mw + mt * 16 + r + rofs;
                C[(size_t)row * N + col] = acc[mt][nt][r] * scale + bb;
            }
        }
    }
}

// ---------------------------------------------------------------------------
// Row-wise softmax in place over x[rows, N], one block per row
// ---------------------------------------------------------------------------
__global__ __launch_bounds__(256)
void softmax_rows(float* __restrict__ x, int N) {
    __shared__ float red[256];
    float* p = x + (size_t)blockIdx.x * N;
    const int tid = threadIdx.x;

    float m = -INFINITY;
    for (int j = tid; j < N; j += 256) m = fmaxf(m, p[j]);
    red[tid] = m; __syncthreads();
    for (int s = 128; s > 0; s >>= 1) {
        if (tid < s) red[tid] = fmaxf(red[tid], red[tid + s]);
        __syncthreads();
    }
    m = red[0]; __syncthreads();

    float sum = 0.0f;
    for (int j = tid; j < N; j += 256) {
        float e = __expf(p[j] - m);
        p[j] = e;
        sum += e;
    }
    red[tid] = sum; __syncthreads();
    for (int s = 128; s > 0; s >>= 1) {
        if (tid < s) red[tid] += red[tid + s];
        __syncthreads();
    }
    const float inv = 1.0f / red[0];
    for (int j = tid; j < N; j += 256) p[j] *= inv;
}

// ---------------------------------------------------------------------------
// 3-phase chunked column scan: cs[0,:]=0; cs[t+1,:]=cs[t,:]+x[t,:]
// ---------------------------------------------------------------------------
__global__ __launch_bounds__(256)
void col_partial(const float* __restrict__ x, float* __restrict__ partial,
                 int D, int CH) {
    const int c  = blockIdx.x * 256 + threadIdx.x;
    const int ch = blockIdx.y;
    const int t0 = ch * CH;
    float s = 0.0f;
    for (int t = t0; t < t0 + CH; ++t) s += x[(size_t)t * D + c];
    partial[(size_t)ch * D + c] = s;
}

__global__ __launch_bounds__(256)
void col_scan_partials(float* __restrict__ partial, int D, int nch) {
    const int c = blockIdx.x * 256 + threadIdx.x;
    float run = 0.0f;
    for (int ch = 0; ch < nch; ++ch) {
        float t = partial[(size_t)ch * D + c];
        partial[(size_t)ch * D + c] = run;
        run += t;
    }
}

__global__ __launch_bounds__(256)
void col_cumsum_chunks(const float* __restrict__ x,
                       const float* __restrict__ partial,
                       float* __restrict__ cs, int D, int CH) {
    const int c  = blockIdx.x * 256 + threadIdx.x;
    const int ch = blockIdx.y;
    const int t0 = ch * CH;
    float run = partial[(size_t)ch * D + c];
    if (ch == 0) cs[c] = 0.0f;
    for (int t = t0; t < t0 + CH; ++t) {
        run += x[(size_t)t * D + c];
        cs[(size_t)(t + 1) * D + c] = run;
    }
}

// ---------------------------------------------------------------------------
// Span feature assembly (float4 path): out row = 3092 floats = 773 float4
//   [cs[end+1]-cs[start] | states[start] | states[end] | dist_embed[bin]]
// ---------------------------------------------------------------------------
__global__ __launch_bounds__(128)
void span_assemble(const float* __restrict__ cs,
                   const float* __restrict__ states,
                   const float* __restrict__ dist_embed,
                   const int* __restrict__ span_starts,
                   const int* __restrict__ span_lengths,
                   float* __restrict__ out, int D) {
    const int s      = blockIdx.x;
    const int start  = span_starts[s];
    const int len    = span_lengths[s];
    const int end    = start + len;
    const int length = len + 1;

    int bin = 0;
    bin += (length > 1);  bin += (length > 2);  bin += (length > 3);
    bin += (length > 4);  bin += (length > 8);  bin += (length > 16);
    bin += (length > 32); bin += (length > 64);

    const int D4 = D >> 2;  // 256
    float4* o4 = (float4*)(out + (size_t)s * (3 * D + 20));
    const float4* csE4 = (const float4*)(cs + (size_t)(end + 1) * D);
    const float4* csS4 = (const float4*)(cs + (size_t)start * D);
    const float4* stS4 = (const float4*)(states + (size_t)start * D);
    const float4* stE4 = (const float4*)(states + (size_t)end * D);

    for (int j = threadIdx.x; j < D4; j += blockDim.x) {
        float4 a = csE4[j], b = csS4[j];
        float4 r;
        r.x = a.x - b.x; r.y = a.y - b.y; r.z = a.z - b.z; r.w = a.w - b.w;
        o4[j]          = r;
        o4[D4 + j]     = stS4[j];
        o4[2 * D4 + j] = stE4[j];
    }
    if (threadIdx.x < 5)   // 20 floats = 5 float4 (bin*20 is 16B aligned)
        o4[3 * D4 + threadIdx.x] =
            ((const float4*)(dist_embed + bin * 20))[threadIdx.x];
}

// ---------------------------------------------------------------------------
extern "C" void kernel_launch(void* const* d_in, const int* in_sizes, int n_in,
                              void* d_out, int out_size, void* d_ws,
                              size_t ws_size, hipStream_t stream) {
    (void)in_sizes; (void)n_in; (void)out_size; (void)ws_size;

    constexpr int T = 4096, D = 1024, S = 32768;
    constexpr int CH = 128, NCH = T / CH;   // cumsum chunking
    const float inv_sqrt_d = 0.03125f;      // 1/sqrt(1024)

    const float* embeds = (const float*)d_in[0];
    const float* states = (const float*)d_in[1];
    const float* Wq = (const float*)d_in[2];
    const float* bq = (const float*)d_in[3];
    const float* Wk = (const float*)d_in[4];
    const float* bk = (const float*)d_in[5];
    const float* Wv = (const float*)d_in[6];
    const float* bv = (const float*)d_in[7];
    const float* Wo = (const float*)d_in[8];
    const float* bo = (const float*)d_in[9];
    const float* dist_embed   = (const float*)d_in[10];
    const int*   span_starts  = (const int*)d_in[11];
    const int*   span_lengths = (const int*)d_in[12];
    float* out = (float*)d_out;

    float* ws = (float*)d_ws;
    const size_t TD = (size_t)T * D;          // 4M floats
    float* q        = ws;                     // [T,D]
    float* k        = ws + TD;                // [T,D]
    float* v        = ws + 2 * TD;            // [T,D]
    float* scores   = ws + 3 * TD;            // [T,T] (64 MB)
    float* attn_out = ws;                     // reuse q   (dead after scores)
    float* attn_emb = ws + TD;                // reuse k   (dead after scores)
    float* cs       = ws + 3 * TD;            // reuse scores (dead after attn_out)
    float* partial  = ws + 3 * TD + 8 * TD / 4;  // 32x1024 scan partials

    const dim3 blk(128);
    const dim3 gProj(D / 64, T / 128);   // 16 x 32
    const dim3 gScore(T / 64, T / 128);  // 64 x 32

    // 1-3: Q/K/V projections (x @ W^T + b), fp32 WMMA
    gemm128_wmma<true><<<gProj, blk, 0, stream>>>(states, Wq, bq, q, T, D, D, 1.0f);
    gemm128_wmma<true><<<gProj, blk, 0, stream>>>(states, Wk, bk, k, T, D, D, 1.0f);
    gemm128_wmma<true><<<gProj, blk, 0, stream>>>(embeds, Wv, bv, v, T, D, D, 1.0f);

    // 4: scores = (q @ k^T) / sqrt(D)   [T,T]
    gemm128_wmma<true><<<gScore, blk, 0, stream>>>(q, k, nullptr, scores, T, T, D,
                                                   inv_sqrt_d);
    // 5: softmax per row
    softmax_rows<<<T, 256, 0, stream>>>(scores, T);

    // 6: attn_out = attn @ v            [T,D]  (B non-transposed)
    gemm128_wmma<false><<<gProj, blk, 0, stream>>>(scores, v, nullptr, attn_out,
                                                   T, D, T, 1.0f);
    // 7: attn_emb = attn_out @ Wo^T + bo
    gemm128_wmma<true><<<gProj, blk, 0, stream>>>(attn_out, Wo, bo, attn_emb,
                                                  T, D, D, 1.0f);

    // 8: column prefix sums cs[T+1, D] (3-phase chunked scan)
    col_partial<<<dim3(D / 256, NCH), dim3(256), 0, stream>>>(attn_emb, partial, D, CH);
    col_scan_partials<<<dim3(D / 256), dim3(256), 0, stream>>>(partial, D, NCH);
    col_cumsum_chunks<<<dim3(D / 256, NCH), dim3(256), 0, stream>>>(attn_emb, partial,
                                                                    cs, D, CH);

    // 9: span feature assembly (float4, writes the 405 MB output)
    span_assemble<<<S, 128, 0, stream>>>(cs, states, dist_embed, span_starts,
                                         span_lengths, out, D);
}